// HistLayer_50998441673272
// MI455X (gfx1250) — compile-verified
//
#include <hip/hip_runtime.h>
#include <hip/hip_bf16.h>

// HistLayer: 128-bin soft histogram over 24 channel-images of 256x256 fp32.
// Memory-bound (6.3 MB in @ 23.3 TB/s ~ 0.27us). Strategy:
//   - async global->LDS staging (double buffered), gfx1250 ASYNCcnt path
//   - 32 LDS histogram copies (one per wave32 lane) -> conflict-free ds_add_f32
//   - 32->1 copy reduction via chained V_WMMA_F32_16X16X4_F32 with A = ones
//   - global atomicAdd f32 merges the 8 splits per channel into d_out

#define NUM_BINS   128
#define BLOCK      256
#define SPLIT      8          // workgroups per channel-image
#define TILE_FLOATS 1024      // BLOCK threads * float4
#define TILES      8          // tiles per workgroup: 8*1024*8 = 65536 px/channel
#define CH_PIXELS  65536
#define NCHAN      24

#if defined(__has_builtin)
#if __has_builtin(__builtin_amdgcn_global_load_async_to_lds_b128) && \
    __has_builtin(__builtin_amdgcn_s_wait_asynccnt)
#define USE_ASYNC 1
#endif
#if __has_builtin(__builtin_amdgcn_wmma_f32_16x16x4_f32)
#define USE_WMMA44 1
#endif
#endif

typedef __attribute__((ext_vector_type(2))) float v2f;
typedef __attribute__((ext_vector_type(4))) float v4f;
typedef __attribute__((ext_vector_type(8))) float v8f;
typedef __attribute__((ext_vector_type(4))) int   v4i;

// builtin signature (from round-1/2 diagnostics):
//   global_load_async_to_lds_b128(AS1 v4i* src, AS3 v4i* dst, imm offset, imm cpol)
typedef __attribute__((address_space(1))) v4i* gbl_v4i_p;
typedef __attribute__((address_space(3))) v4i* lds_v4i_p;

__global__ __launch_bounds__(BLOCK) void hist_zero_kernel(float* __restrict__ out, int n) {
    int i = blockIdx.x * BLOCK + threadIdx.x;
    if (i < n) out[i] = 0.0f;
}

__global__ __launch_bounds__(BLOCK) void hist_kernel(const float* __restrict__ in,
                                                     float* __restrict__ out) {
    __shared__ float stage[2][TILE_FLOATS];      // 8 KB double buffer
    __shared__ float hist[NUM_BINS][32];         // 16 KB: 32 copies, one per lane

    const int tid  = threadIdx.x;
    const int lane = tid & 31;                   // wave32
    const int ch   = blockIdx.x / SPLIT;         // 0..23
    const int part = blockIdx.x % SPLIT;
    const int base = ch * CH_PIXELS + part * (TILES * TILE_FLOATS);

    // zero the LDS histogram copies
    for (int i = tid; i < NUM_BINS * 32; i += BLOCK) (&hist[0][0])[i] = 0.0f;
    __syncthreads();

    // per-pixel soft-bin: y = 1.01^(1/256 - |x - c_bin|), counted iff y > 1
    auto proc = [&](float x) {
        int bin = (int)(x * 128.0f);
        bin = bin < 127 ? bin : 127;
        bin = bin > 0 ? bin : 0;
        float c = (float)(2 * bin + 1) * (1.0f / 256.0f);     // exact in fp32
        float d = 0.00390625f - __builtin_fabsf(x - c);       // 1/256 - |x-c|
        float y = __builtin_exp2f(d * 0.014355292977070041f); // log2(1.01)
        if (y > 1.0f) atomicAdd(&hist[bin][lane], y);         // ds_add_f32, no lane conflicts
    };

#ifdef USE_ASYNC
    // ---- async global -> LDS streaming, double buffered ----
    __builtin_amdgcn_global_load_async_to_lds_b128(
        (gbl_v4i_p)(in + base + tid * 4),
        (lds_v4i_p)(&stage[0][tid * 4]), 0, 0);
#pragma unroll
    for (int t = 0; t < TILES; ++t) {
        if (t + 1 < TILES) {
            __builtin_amdgcn_global_load_async_to_lds_b128(
                (gbl_v4i_p)(in + base + (t + 1) * TILE_FLOATS + tid * 4),
                (lds_v4i_p)(&stage[(t + 1) & 1][tid * 4]), 0, 0);
            __builtin_amdgcn_s_wait_asynccnt(1);   // current tile's load done
        } else {
            __builtin_amdgcn_s_wait_asynccnt(0);
        }
        const v4f v = *reinterpret_cast<const v4f*>(&stage[t & 1][tid * 4]);
        proc(v.x); proc(v.y); proc(v.z); proc(v.w);
    }
#else
    // fallback: direct vectorized global loads
#pragma unroll
    for (int t = 0; t < TILES; ++t) {
        const v4f v = *reinterpret_cast<const v4f*>(in + base + t * TILE_FLOATS + tid * 4);
        proc(v.x); proc(v.y); proc(v.z); proc(v.w);
    }
#endif

    __syncthreads();  // all ds_add_f32 complete; hist visible to all waves

#ifdef USE_WMMA44
    // ---- reduce 32 copies per bin with chained f32 WMMAs ----
    // Wave w owns bins [16w, 16w+16). A = ones(16x4) so D[m][n] = sum_k B[k][n];
    // with A all-ones the K<->lane mapping of B is irrelevant (any row permutation
    // of B yields the same column sums). 8 chained WMMAs accumulate all 32 copies.
    {
        const int wv      = tid >> 5;       // 0..7
        const int n       = lane & 15;      // bin within group / column
        const int h       = lane >> 4;      // lane half selects 2 of the 4 K rows
        const int binbase = wv * 16;
        const v2f a = {1.0f, 1.0f};
        v8f acc = {};
#pragma unroll
        for (int k = 0; k < 8; ++k) {
            v2f b;
            b.x = hist[binbase + n][4 * k + 2 * h];
            b.y = hist[binbase + n][4 * k + 2 * h + 1];
            acc = __builtin_amdgcn_wmma_f32_16x16x4_f32(
                false, a, false, b, (short)0, acc, false, false);
        }
        if (lane < 16) {
            // D row 0, column n lives in acc[0] of lanes 0..15
            atomicAdd(&out[ch * NUM_BINS + binbase + n], acc[0] * (1.0f / 65536.0f));
        }
    }
#else
    if (tid < NUM_BINS) {
        float s = 0.0f;
#pragma unroll
        for (int c2 = 0; c2 < 32; ++c2) s += hist[tid][c2];
        atomicAdd(&out[ch * NUM_BINS + tid], s * (1.0f / 65536.0f));
    }
#endif
}

extern "C" void kernel_launch(void* const* d_in, const int* in_sizes, int n_in,
                              void* d_out, int out_size, void* d_ws, size_t ws_size,
                              hipStream_t stream) {
    const float* in = (const float*)d_in[0];   // [8,3,256,256] fp32
    float* out = (float*)d_out;                // [8, 3*128] fp32

    hist_zero_kernel<<<(out_size + BLOCK - 1) / BLOCK, BLOCK, 0, stream>>>(out, out_size);
    hist_kernel<<<NCHAN * SPLIT, BLOCK, 0, stream>>>(in, out);
}